// LieAdvection_26173530702669
// MI455X (gfx1250) — compile-verified
//
#include <hip/hip_runtime.h>
#include <stdint.h>

// LieAdvection (RK2 flow warp + bilinear border gather) for MI455X / gfx1250.
//
// Memory-bound: ~270 MB of traffic -> ~12 us floor at 23.3 TB/s HBM.
// WMMA is not applicable (the K=4 bilinear blend has a per-pixel B matrix,
// which cannot be shared across the WMMA M dimension), so the CDNA5-specific
// paths used are the async tensor-memory pipeline
// (global_load_async_to_lds_b128 + s_wait_asynccnt) for staging flow rows,
// wave32-aware lane mapping (each wave stores 512B fully contiguous), b128
// vector loads on the contiguous r dimension, and non-temporal stores so the
// streaming 128 MiB output does not evict the reusable h_prev from the
// 192 MB L2.

#define B_ 4
#define C_ 16
#define H_ 128
#define W_ 128
#define R_ 32
#define G_ 4

typedef __attribute__((ext_vector_type(4))) float f4;   // native vector (b128)

__device__ __forceinline__ void async_gld_b128_to_lds(uint32_t lds_off, uint64_t gaddr) {
  // CDNA5 async global->LDS DMA path (tracked by ASYNCcnt, no VGPR staging).
  asm volatile("global_load_async_to_lds_b128 %0, %1, off"
               :: "v"(lds_off), "v"(gaddr) : "memory");
}

__device__ __forceinline__ void wait_asynccnt0() {
#if __has_builtin(__builtin_amdgcn_s_wait_asynccnt)
  __builtin_amdgcn_s_wait_asynccnt(0);
#else
  asm volatile("s_wait_asynccnt 0" ::: "memory");
#endif
}

__device__ __forceinline__ f4 lerp4(f4 a, f4 b, float t) {
  return a + t * (b - a);   // component-wise, scalar t broadcast
}

__global__ __launch_bounds__(256) void lie_advect_kernel(
    const float* __restrict__ h_prev,   // [B,C,H,W,R]
    const float* __restrict__ flows,    // [B,G,2,H,W]
    const float* __restrict__ dt,       // [B]
    float* __restrict__ out)            // [B,C,H,W,R]
{
  const int bh  = blockIdx.x;          // one block per (b, h) row
  const int b   = bh >> 7;             // / H_
  const int h   = bh & (H_ - 1);
  const int tid = threadIdx.x;

  __shared__ float sK1[G_ * 2 * W_];   // k1 rows: [g][ch][w]   (4 KiB)
  __shared__ int   sX0[G_ * W_];       // per-(g,w) sample corner + weights
  __shared__ int   sY0[G_ * W_];
  __shared__ float sWx[G_ * W_];
  __shared__ float sWy[G_ * W_];

  // ---- Stage 0: async DMA the 8 flow rows (k1 for 4 groups x 2 ch) to LDS.
  {
    const int li = tid << 2;                     // 4 floats / thread, 1024 total
    const int g  = li >> 8;
    const int ch = (li >> 7) & 1;
    const int w  = li & (W_ - 1);
    const float* gsrc = flows + (((b * G_ + g) * 2 + ch) * H_ + h) * W_ + w;
    async_gld_b128_to_lds((uint32_t)(uintptr_t)&sK1[li], (uint64_t)(uintptr_t)gsrc);
  }
  wait_asynccnt0();
  __syncthreads();

  const float dtb = dt[b];

  // ---- Stage 1: RK2 sample coordinate per (g, w); 512 items, 2 per thread.
  for (int it = tid; it < G_ * W_; it += 256) {
    const int g = it >> 7;
    const int w = it & (W_ - 1);
    const float bx = -1.0f + (float)w * (2.0f / (float)(W_ - 1));
    const float by = -1.0f + (float)h * (2.0f / (float)(H_ - 1));
    const float k1x = sK1[(g * 2 + 0) * W_ + w];
    const float k1y = sK1[(g * 2 + 1) * W_ + w];

    // midpoint grid -> pixel coords (align_corners=False, border clamp)
    const float gmx = bx - 0.5f * k1x * dtb;
    const float gmy = by - 0.5f * k1y * dtb;
    float px = fminf(fmaxf((gmx + 1.0f) * (0.5f * W_) - 0.5f, 0.0f), (float)(W_ - 1));
    float py = fminf(fmaxf((gmy + 1.0f) * (0.5f * H_) - 0.5f, 0.0f), (float)(H_ - 1));
    float fx0 = floorf(px), fy0 = floorf(py);
    int x0 = (int)fx0, y0 = (int)fy0;
    int x1 = min(x0 + 1, W_ - 1), y1 = min(y0 + 1, H_ - 1);
    float wx = px - fx0, wy = py - fy0;

    // k2 = bilinear sample of both flow channels at midpoint (L2-hot, 2 MiB)
    const float* fX = flows + ((b * G_ + g) * 2) * (H_ * W_);
    const float* fY = fX + H_ * W_;
    const int i00 = y0 * W_ + x0, i01 = y0 * W_ + x1;
    const int i10 = y1 * W_ + x0, i11 = y1 * W_ + x1;
    float t0 = fX[i00] + wx * (fX[i01] - fX[i00]);
    float t1 = fX[i10] + wx * (fX[i11] - fX[i10]);
    const float k2x = t0 + wy * (t1 - t0);
    t0 = fY[i00] + wx * (fY[i01] - fY[i00]);
    t1 = fY[i10] + wx * (fY[i11] - fY[i10]);
    const float k2y = t0 + wy * (t1 - t0);

    // final grid -> pixel coords
    const float gx2 = bx - k2x * dtb;
    const float gy2 = by - k2y * dtb;
    float qx = fminf(fmaxf((gx2 + 1.0f) * (0.5f * W_) - 0.5f, 0.0f), (float)(W_ - 1));
    float qy = fminf(fmaxf((gy2 + 1.0f) * (0.5f * H_) - 0.5f, 0.0f), (float)(H_ - 1));
    const float qx0 = floorf(qx), qy0 = floorf(qy);
    sX0[it] = (int)qx0;
    sY0[it] = (int)qy0;
    sWx[it] = qx - qx0;
    sWy[it] = qy - qy0;
  }
  __syncthreads();

  // ---- Stage 2: gather 512 values (16 c x 32 r) per pixel with b128 vector accesses.
  // Lane mapping: pix = lane>>3 (4 consecutive pixels per wave),
  //               r0  = 4*(lane&7) (float4 run of r), group g = r0>>3.
  // Wave store = 4 pixels x 128B = 512B fully contiguous (nontemporal b128).
  const int lane = tid & 31;
  const int wv   = tid >> 5;          // 8 waves
  const int pix  = lane >> 3;         // which of the 4 pixels in the quad
  const int sub  = lane & 7;
  const int r0   = sub << 2;          // r .. r+3
  const int gl   = sub >> 1;          // group of this r-run
  const int bOff = b * (C_ * H_ * W_ * R_);
  const int oRow = ((b * C_) * H_ + h) * (W_ * R_);   // + c*H*W*R in loop

#pragma unroll 2
  for (int q = wv; q < (C_ * W_) / 4; q += 8) {   // 512 pixel-quads
    const int c = q >> 5;                         // 32 quads per channel
    const int w = ((q & 31) << 2) + pix;
    const int idx = (gl << 7) | w;
    const int x0 = sX0[idx], y0 = sY0[idx];
    const float wx = sWx[idx], wy = sWy[idx];
    const int x1 = min(x0 + 1, W_ - 1), y1 = min(y0 + 1, H_ - 1);

    const float* cb = h_prev + bOff + c * (H_ * W_ * R_) + r0;
    const f4 v00 = *(const f4*)(cb + (y0 * W_ + x0) * R_);
    const f4 v01 = *(const f4*)(cb + (y0 * W_ + x1) * R_);
    const f4 v10 = *(const f4*)(cb + (y1 * W_ + x0) * R_);
    const f4 v11 = *(const f4*)(cb + (y1 * W_ + x1) * R_);

    const f4 t0  = lerp4(v00, v01, wx);
    const f4 t1  = lerp4(v10, v11, wx);
    const f4 val = lerp4(t0, t1, wy);

    // streaming output: NT store keeps h_prev resident in L2
    f4* dst = (f4*)(out + oRow + c * (H_ * W_ * R_) + w * R_ + r0);
    __builtin_nontemporal_store(val, dst);
  }
}

extern "C" void kernel_launch(void* const* d_in, const int* in_sizes, int n_in,
                              void* d_out, int out_size, void* d_ws, size_t ws_size,
                              hipStream_t stream) {
  (void)in_sizes; (void)n_in; (void)d_ws; (void)ws_size; (void)out_size;
  const float* h_prev = (const float*)d_in[0];
  const float* flows  = (const float*)d_in[1];
  const float* dtv    = (const float*)d_in[2];
  float* out = (float*)d_out;

  dim3 grid(B_ * H_);   // 512 blocks: one per (b, h)
  dim3 block(256);      // 8 wave32 waves
  hipLaunchKernelGGL(lie_advect_kernel, grid, block, 0, stream,
                     h_prev, flows, dtv, out);
}